// NUFNO2d_57140244906258
// MI455X (gfx1250) — compile-verified
//
#include <hip/hip_runtime.h>
#include <math.h>

// ---------------- problem constants (match reference) ----------------
#define B_   4
#define NX   4096
#define NY   8192
#define G_   4096
#define HX_  64
#define HY_  64
#define WID_ 32
#define M1_  8
#define M2_  8
#define S_   16
#define EPS_ 1e-6f
#define KSPLIT 4

#define TWO_PI_F 6.28318530717958647692f
#define PI_F     3.14159265358979323846f

typedef __attribute__((ext_vector_type(2))) float v2f;
typedef __attribute__((ext_vector_type(8))) float v8f;

// V_WMMA_F32_16X16X4_F32 : A 16x4 f32 (2 VGPR), B 4x16 f32 (2 VGPR), C/D 16x16 f32 (8 VGPR)
__device__ __forceinline__ v8f wmma4(v2f a, v2f b, v8f c) {
  return __builtin_amdgcn_wmma_f32_16x16x4_f32(false, a, false, b, (short)0, c,
                                               false, false);
}

// streaming (non-temporal) 64-bit load for read-once weight matrices
__device__ __forceinline__ v2f ldnt2(const float* p) {
  return __builtin_nontemporal_load((const v2f*)p);
}

__device__ __forceinline__ float gelu_exact(float x) {
  return 0.5f * x * (1.0f + erff(x * 0.70710678118654752440f));
}

// ---------------- workspace layout (float elements) ----------------
constexpr size_t OFF_PXE  = 0;                                        // KSPLIT*B*G
constexpr size_t OFF_PLE  = OFF_PXE + (size_t)KSPLIT * B_ * G_;       // KSPLIT*B*G*2
constexpr size_t OFF_H    = OFF_PLE + (size_t)KSPLIT * B_ * G_ * 2;   // B*WID*G
constexpr size_t OFF_HB   = OFF_H + (size_t)B_ * WID_ * G_;           // B*WID*G (h1)
constexpr size_t OFF_ST   = OFF_HB + (size_t)B_ * WID_ * G_;          // B*WID*64*9*2
constexpr size_t OFF_FR   = OFF_ST + (size_t)B_ * WID_ * HX_ * 9 * 2; // B*WID*16*8*2
constexpr size_t OFF_FO   = OFF_FR + (size_t)B_ * WID_ * 16 * 8 * 2;  // B*WID*16*8*2
constexpr size_t OFF_NFT  = OFF_FO + (size_t)B_ * WID_ * 16 * 8 * 2;  // B*WID*17*9*2
constexpr size_t OFF_IMG  = OFF_NFT + (size_t)B_ * WID_ * 17 * 9 * 2; // B*16*289*2
constexpr size_t OFF_HL   = OFF_IMG + (size_t)B_ * 16 * 289 * 2;      // B*G
constexpr size_t OFF_XNU  = OFF_HL + (size_t)B_ * G_;                 // B*NY
constexpr size_t OFF_MNMX = OFF_XNU + (size_t)B_ * NY;                // B*S*4 (int bits)
constexpr size_t OFF_SEG  = OFF_MNMX + (size_t)B_ * S_ * 4;           // B*NY (int)
constexpr size_t OFF_PDE  = OFF_SEG + (size_t)B_ * NY;                // KSPLIT*B*NY

// =====================================================================
// Skinny WMMA GEMM, K split 4 ways, software pipelined (1 group in flight).
// pxe[c][b][g] = sum_{k in chunk} x[b,k]*W[g,k]
// =====================================================================
__global__ void k_gemm_xe(const float* __restrict__ x, const float* __restrict__ W,
                          float* __restrict__ pxe) {
  const int wave = (blockIdx.x * blockDim.x + threadIdx.x) >> 5;
  const int lane = threadIdx.x & 31;
  const int tile = wave >> 2;
  const int chunk = wave & 3;
  const int g0 = tile * 16;
  const int m = lane & 15;
  const int klo = (lane >> 4) << 1; // 0 or 2
  const int kbeg = chunk * (NX / KSPLIT);
  const int kend = kbeg + (NX / KSPLIT);
  const float* Arow = W + (size_t)(g0 + m) * NX + klo;
  const int msafe = (m < B_) ? m : 0;
  const float* Brow = x + (size_t)msafe * NX + klo;
  const bool act = (m < B_);
  v8f acc = {};
  v2f a0[4], b0[4], a1[4], b1[4];
#pragma unroll
  for (int u = 0; u < 4; ++u) {
    a0[u] = ldnt2(Arow + kbeg + u * 4);
    b0[u] = *(const v2f*)(Brow + kbeg + u * 4);
  }
  for (int k0 = kbeg; k0 + 16 < kend; k0 += 16) {
#pragma unroll
    for (int u = 0; u < 4; ++u) {
      a1[u] = ldnt2(Arow + k0 + 16 + u * 4);
      b1[u] = *(const v2f*)(Brow + k0 + 16 + u * 4);
    }
    __builtin_prefetch(Arow + k0 + 4096, 0, 0);
#pragma unroll
    for (int u = 0; u < 4; ++u) {
      v2f bb; bb.x = act ? b0[u].x : 0.0f; bb.y = act ? b0[u].y : 0.0f;
      acc = wmma4(a0[u], bb, acc);
    }
#pragma unroll
    for (int u = 0; u < 4; ++u) { a0[u] = a1[u]; b0[u] = b1[u]; }
  }
#pragma unroll
  for (int u = 0; u < 4; ++u) {
    v2f bb; bb.x = act ? b0[u].x : 0.0f; bb.y = act ? b0[u].y : 0.0f;
    acc = wmma4(a0[u], bb, acc);
  }
  const int nb = lane & 15;
  const int mb = (lane >> 4) * 8;
  if (nb < B_) {
#pragma unroll
    for (int v = 0; v < 8; ++v)
      pxe[((size_t)(chunk * B_ + nb)) * G_ + g0 + mb + v] = acc[v];
  }
}

// ple[c][b][g][ch] partials of le[b,g,ch] = sum_n loc[b,n,ch] * W_embed2[g,n]
__global__ void k_gemm_le(const float* __restrict__ loc, const float* __restrict__ W,
                          float* __restrict__ ple) {
  const int wave = (blockIdx.x * blockDim.x + threadIdx.x) >> 5;
  const int lane = threadIdx.x & 31;
  const int tile = wave >> 2;
  const int chunk = wave & 3;
  const int g0 = tile * 16;
  const int m = lane & 15;
  const int klo = (lane >> 4) << 1;
  const int kbeg = chunk * (NY / KSPLIT);
  const int kend = kbeg + (NY / KSPLIT);
  const float* Arow = W + (size_t)(g0 + m) * NY + klo;
  const bool act = (m < 2 * B_);
  const int bb_ = act ? (m >> 1) : 0;
  const int cc = act ? (m & 1) : 0;
  const float* Bbase = loc + ((size_t)bb_ * NY + klo) * 2 + cc; // element k -> Bbase[2k]
  v8f acc = {};
  v2f a0[4], b0[4], a1[4], b1[4];
#pragma unroll
  for (int u = 0; u < 4; ++u) {
    int k = kbeg + u * 4;
    a0[u] = ldnt2(Arow + k);
    b0[u].x = Bbase[(size_t)k * 2];
    b0[u].y = Bbase[(size_t)(k + 1) * 2];
  }
  for (int k0 = kbeg; k0 + 16 < kend; k0 += 16) {
#pragma unroll
    for (int u = 0; u < 4; ++u) {
      int k = k0 + 16 + u * 4;
      a1[u] = ldnt2(Arow + k);
      b1[u].x = Bbase[(size_t)k * 2];
      b1[u].y = Bbase[(size_t)(k + 1) * 2];
    }
    __builtin_prefetch(Arow + k0 + 4096, 0, 0);
#pragma unroll
    for (int u = 0; u < 4; ++u) {
      v2f bv; bv.x = act ? b0[u].x : 0.0f; bv.y = act ? b0[u].y : 0.0f;
      acc = wmma4(a0[u], bv, acc);
    }
#pragma unroll
    for (int u = 0; u < 4; ++u) { a0[u] = a1[u]; b0[u] = b1[u]; }
  }
#pragma unroll
  for (int u = 0; u < 4; ++u) {
    v2f bv; bv.x = act ? b0[u].x : 0.0f; bv.y = act ? b0[u].y : 0.0f;
    acc = wmma4(a0[u], bv, acc);
  }
  const int n = lane & 15;
  const int mb = (lane >> 4) * 8;
  if (n < 2 * B_) {
    const int ob = n >> 1, oc = n & 1;
#pragma unroll
    for (int v = 0; v < 8; ++v)
      ple[(((size_t)(chunk * B_ + ob)) * G_ + g0 + mb + v) * 2 + oc] = acc[v];
  }
}

// pde[c][b][y] partials of sum_g hl[b,g]*W_de3[y,g]
__global__ void k_gemm_de3(const float* __restrict__ hl, const float* __restrict__ W,
                           float* __restrict__ pde) {
  const int wave = (blockIdx.x * blockDim.x + threadIdx.x) >> 5;
  const int lane = threadIdx.x & 31;
  const int tile = wave >> 2;
  const int chunk = wave & 3;
  const int y0 = tile * 16;
  const int m = lane & 15;
  const int klo = (lane >> 4) << 1;
  const int kbeg = chunk * (G_ / KSPLIT);
  const int kend = kbeg + (G_ / KSPLIT);
  const float* Arow = W + (size_t)(y0 + m) * G_ + klo;
  const int msafe = (m < B_) ? m : 0;
  const float* Brow = hl + (size_t)msafe * G_ + klo;
  const bool act = (m < B_);
  v8f acc = {};
  v2f a0[4], b0[4], a1[4], b1[4];
#pragma unroll
  for (int u = 0; u < 4; ++u) {
    a0[u] = ldnt2(Arow + kbeg + u * 4);
    b0[u] = *(const v2f*)(Brow + kbeg + u * 4);
  }
  for (int k0 = kbeg; k0 + 16 < kend; k0 += 16) {
#pragma unroll
    for (int u = 0; u < 4; ++u) {
      a1[u] = ldnt2(Arow + k0 + 16 + u * 4);
      b1[u] = *(const v2f*)(Brow + k0 + 16 + u * 4);
    }
    __builtin_prefetch(Arow + k0 + 4096, 0, 0);
#pragma unroll
    for (int u = 0; u < 4; ++u) {
      v2f bv; bv.x = act ? b0[u].x : 0.0f; bv.y = act ? b0[u].y : 0.0f;
      acc = wmma4(a0[u], bv, acc);
    }
#pragma unroll
    for (int u = 0; u < 4; ++u) { a0[u] = a1[u]; b0[u] = b1[u]; }
  }
#pragma unroll
  for (int u = 0; u < 4; ++u) {
    v2f bv; bv.x = act ? b0[u].x : 0.0f; bv.y = act ? b0[u].y : 0.0f;
    acc = wmma4(a0[u], bv, acc);
  }
  const int nb = lane & 15;
  const int mb = (lane >> 4) * 8;
  if (nb < B_) {
#pragma unroll
    for (int v = 0; v < 8; ++v)
      pde[((size_t)(chunk * B_ + nb)) * NY + y0 + mb + v] = acc[v];
  }
}

// reduce de3 partials + bias + NU branch -> final output (B, Ny, 1)
__global__ void k_red_de3(const float* __restrict__ pde, const float* __restrict__ bias,
                          const float* __restrict__ xnu, float* __restrict__ out) {
  int idx = blockIdx.x * blockDim.x + threadIdx.x;
  if (idx >= B_ * NY) return;
  int b = idx / NY, y = idx - b * NY;
  float s = bias[y] + xnu[idx];
#pragma unroll
  for (int c = 0; c < KSPLIT; ++c) s += pde[((size_t)(c * B_ + b)) * NY + y];
  out[idx] = s;
}

// =====================================================================
// fc0 (reduces embed partials): h[b,c,g] = W_fc0[c,:].{xe,le0,le1}+b
// =====================================================================
__global__ void k_fc0(const float* __restrict__ pxe, const float* __restrict__ ple,
                      const float* __restrict__ be1, const float* __restrict__ be2,
                      const float* __restrict__ Wf, const float* __restrict__ bf,
                      float* __restrict__ h) {
  int idx = blockIdx.x * blockDim.x + threadIdx.x;
  if (idx >= B_ * G_) return;
  int b = idx / G_, g = idx - b * G_;
  float f0 = be1[g], f1 = be2[g], f2 = be2[g];
#pragma unroll
  for (int c = 0; c < KSPLIT; ++c) {
    f0 += pxe[((size_t)(c * B_ + b)) * G_ + g];
    f1 += ple[(((size_t)(c * B_ + b)) * G_ + g) * 2 + 0];
    f2 += ple[(((size_t)(c * B_ + b)) * G_ + g) * 2 + 1];
  }
#pragma unroll
  for (int c = 0; c < WID_; ++c) {
    float v = Wf[c * 3 + 0] * f0 + Wf[c * 3 + 1] * f1 + Wf[c * 3 + 2] * f2 + bf[c];
    h[((size_t)(b * WID_ + c)) * G_ + g] = v;
  }
}

// =====================================================================
// FNO spectral conv: separable partial DFT (16 x-modes, 8 y-modes)
// =====================================================================
__global__ void k_dfty(const float* __restrict__ h, float* __restrict__ st) {
  int idx = blockIdx.x * blockDim.x + threadIdx.x;
  if (idx >= B_ * WID_ * HX_ * M1_) return;
  int ky = idx & 7;
  int x = (idx >> 3) & 63;
  int bc = idx >> 9;
  const float* row = h + ((size_t)bc * G_) + x * HY_;
  float th = -TWO_PI_F * (float)ky / 64.0f;
  float cr, sr; __sincosf(th, &sr, &cr);
  float c0 = 1.0f, s0 = 0.0f, ar = 0.0f, ai = 0.0f;
  for (int y = 0; y < HY_; ++y) {
    float v = row[y];
    ar = fmaf(v, c0, ar); ai = fmaf(v, s0, ai);
    float nc = c0 * cr - s0 * sr;
    s0 = c0 * sr + s0 * cr; c0 = nc;
  }
  st[(size_t)idx * 2] = ar; st[(size_t)idx * 2 + 1] = ai;
}

__global__ void k_dftx(const float* __restrict__ st, float* __restrict__ fr) {
  int idx = blockIdx.x * blockDim.x + threadIdx.x;
  if (idx >= B_ * WID_ * 16 * M1_) return;
  int ky = idx & 7;
  int kx = (idx >> 3) & 15;
  int bc = idx >> 7;
  int kxp = (kx < 8) ? kx : kx + 48;
  float th = -TWO_PI_F * (float)kxp / 64.0f;
  float cr, sr; __sincosf(th, &sr, &cr);
  float c0 = 1.0f, s0 = 0.0f, ar = 0.0f, ai = 0.0f;
  for (int x = 0; x < HX_; ++x) {
    const float* t = st + (((size_t)bc * HX_ + x) * 8 + ky) * 2;
    float tr = t[0], ti = t[1];
    ar += tr * c0 - ti * s0;
    ai += tr * s0 + ti * c0;
    float nc = c0 * cr - s0 * sr;
    s0 = c0 * sr + s0 * cr; c0 = nc;
  }
  fr[(size_t)idx * 2] = ar; fr[(size_t)idx * 2 + 1] = ai;
}

__global__ void k_mix(const float* __restrict__ fr, const float* __restrict__ cw1,
                      const float* __restrict__ cw2, float* __restrict__ fo, int layer) {
  int idx = blockIdx.x * blockDim.x + threadIdx.x;
  if (idx >= B_ * WID_ * 16 * M1_) return;
  int ky = idx & 7;
  int kx = (idx >> 3) & 15;
  int o = (idx >> 7) & 31;
  int b = idx >> 12;
  const float* wbase = (kx < 8) ? cw1 : cw2;
  int mx = kx & 7;
  float ar = 0.0f, ai = 0.0f;
#pragma unroll 4
  for (int i = 0; i < WID_; ++i) {
    const float* f = fr + ((((size_t)(b * WID_ + i) * 16) + kx) * 8 + ky) * 2;
    const float* w = wbase + (((((size_t)layer * WID_ + i) * WID_ + o) * 8 + mx) * 8 + ky) * 2;
    float xr = f[0], xi = f[1], wr = w[0], wi = w[1];
    ar += xr * wr - xi * wi;
    ai += xr * wi + xi * wr;
  }
  fo[(size_t)idx * 2] = ar; fo[(size_t)idx * 2 + 1] = ai;
}

__global__ void k_idftx(const float* __restrict__ fo, float* __restrict__ st) {
  int idx = blockIdx.x * blockDim.x + threadIdx.x;
  if (idx >= B_ * WID_ * HX_ * M1_) return;
  int ky = idx & 7;
  int x = (idx >> 3) & 63;
  int bo = idx >> 9;
  float ar = 0.0f, ai = 0.0f;
#pragma unroll
  for (int kx = 0; kx < 16; ++kx) {
    int kxp = (kx < 8) ? kx : kx + 48;
    float th = TWO_PI_F * (float)((kxp * x) & 63) / 64.0f;
    float s, c; __sincosf(th, &s, &c);
    const float* f = fo + (((size_t)bo * 16 + kx) * 8 + ky) * 2;
    ar += f[0] * c - f[1] * s;
    ai += f[0] * s + f[1] * c;
  }
  st[(((size_t)bo * HX_ + x) * 8 + ky) * 2] = ar;
  st[(((size_t)bo * HX_ + x) * 8 + ky) * 2 + 1] = ai;
}

__global__ void k_idfty(const float* __restrict__ st, float* __restrict__ h1) {
  int idx = blockIdx.x * blockDim.x + threadIdx.x;
  if (idx >= B_ * WID_ * G_) return;
  int y = idx & 63;
  int x = (idx >> 6) & 63;
  int bo = idx >> 12;
  const float* g = st + ((size_t)bo * HX_ + x) * 8 * 2;
  float acc = g[0];
#pragma unroll
  for (int ky = 1; ky < 8; ++ky) {
    float th = TWO_PI_F * (float)((ky * y) & 63) / 64.0f;
    float s, c; __sincosf(th, &s, &c);
    acc += 2.0f * (g[ky * 2] * c - g[ky * 2 + 1] * s);
  }
  h1[idx] = acc * (1.0f / 4096.0f);
}

// =====================================================================
// Pointwise 32x32 channel GEMM (WMMA, weights staged in LDS via async DMA)
// + h1 + bias + optional GELU
// =====================================================================
__global__ void k_wpt(float* __restrict__ h, const float* __restrict__ h1,
                      const float* __restrict__ wpt, const float* __restrict__ wptb,
                      int layer, int do_gelu) {
  __shared__ float lws[WID_ * WID_];
  const float* wl = wpt + (size_t)layer * WID_ * WID_;
  {
    // async copy of the 4KB weight tile: 256 threads x 16B
    const float* gsrc = wl + threadIdx.x * 4;
    unsigned ldsoff = (unsigned)(uintptr_t)lws + threadIdx.x * 16u;
    asm volatile("global_load_async_to_lds_b128 %0, %1, off\n\t"
                 "s_wait_asynccnt 0x0"
                 :
                 : "v"(ldsoff), "v"(gsrc)
                 : "memory");
  }
  __syncthreads();

  const int wave = (blockIdx.x * blockDim.x + threadIdx.x) >> 5; // 0 .. B*G/16-1
  const int lane = threadIdx.x & 31;
  const int b = wave >> 8;           // 256 tiles per batch
  const int g0 = (wave & 255) * 16;
  const int m = lane & 15;
  const int klo = (lane >> 4) << 1;
  v8f acc0 = {}, acc1 = {};
#pragma unroll
  for (int k0 = 0; k0 < WID_; k0 += 4) {
    int k = k0 + klo;
    v2f bvec;
    bvec.x = h[((size_t)(b * WID_ + k)) * G_ + g0 + m];
    bvec.y = h[((size_t)(b * WID_ + k + 1)) * G_ + g0 + m];
    v2f a0; a0.x = lws[(0 + m) * WID_ + k];  a0.y = lws[(0 + m) * WID_ + k + 1];
    v2f a1; a1.x = lws[(16 + m) * WID_ + k]; a1.y = lws[(16 + m) * WID_ + k + 1];
    acc0 = wmma4(a0, bvec, acc0);
    acc1 = wmma4(a1, bvec, acc1);
  }
  const int n = lane & 15;
  const int mb = (lane >> 4) * 8;
#pragma unroll
  for (int v = 0; v < 8; ++v) {
    int o0 = mb + v;
    int o1 = o0 + 16;
    size_t i0 = ((size_t)(b * WID_ + o0)) * G_ + g0 + n;
    size_t i1 = ((size_t)(b * WID_ + o1)) * G_ + g0 + n;
    float r0 = acc0[v] + h1[i0] + wptb[layer * WID_ + o0];
    float r1 = acc1[v] + h1[i1] + wptb[layer * WID_ + o1];
    if (do_gelu) { r0 = gelu_exact(r0); r1 = gelu_exact(r1); }
    h[i0] = r0; h[i1] = r1;
  }
}

// =====================================================================
// fc1 (32->128, gelu) then fc2 (128->1), per latent point
// =====================================================================
__global__ void k_fc12(const float* __restrict__ h, const float* __restrict__ W1,
                       const float* __restrict__ b1, const float* __restrict__ W2,
                       const float* __restrict__ b2, float* __restrict__ hl) {
  int idx = blockIdx.x * blockDim.x + threadIdx.x;
  if (idx >= B_ * G_) return;
  int b = idx / G_, g = idx - b * G_;
  float hv[WID_];
#pragma unroll
  for (int c = 0; c < WID_; ++c) hv[c] = h[((size_t)(b * WID_ + c)) * G_ + g];
  float acc = 0.0f;
  for (int j = 0; j < 128; ++j) {
    float t = b1[j];
    const float* wr = W1 + j * WID_;
#pragma unroll
    for (int c = 0; c < WID_; ++c) t = fmaf(wr[c], hv[c], t);
    acc = fmaf(W2[j], gelu_exact(t), acc);
  }
  hl[idx] = acc + b2[0];
}

// =====================================================================
// NU spectral path: centered 65x65 ortho DFT window (fx in [-8..8], fy in [-8..0])
// =====================================================================
__global__ void k_nudfty(const float* __restrict__ h, float* __restrict__ st) {
  int idx = blockIdx.x * blockDim.x + threadIdx.x;
  if (idx >= B_ * WID_ * HX_ * 9) return;
  int jy = idx % 9;
  int t = idx / 9;
  int x = t & 63;
  int bc = t >> 6;
  const float* row = h + (size_t)bc * G_ + x * HY_;
  float th = -TWO_PI_F * (float)(jy - 8) / 65.0f;
  float cr, sr; __sincosf(th, &sr, &cr);
  float c0 = 1.0f, s0 = 0.0f, ar = 0.0f, ai = 0.0f;
  for (int y = 0; y < HY_; ++y) {
    float v = row[y];
    ar = fmaf(v, c0, ar); ai = fmaf(v, s0, ai);
    float nc = c0 * cr - s0 * sr;
    s0 = c0 * sr + s0 * cr; c0 = nc;
  }
  st[(size_t)idx * 2] = ar; st[(size_t)idx * 2 + 1] = ai;
}

__global__ void k_nudftx(const float* __restrict__ st, float* __restrict__ nft) {
  int idx = blockIdx.x * blockDim.x + threadIdx.x;
  if (idx >= B_ * WID_ * 17 * 9) return;
  int jy = idx % 9;
  int t = idx / 9;
  int jx = t % 17;
  int bc = t / 17;
  float th = -TWO_PI_F * (float)(jx - 8) / 65.0f;
  float cr, sr; __sincosf(th, &sr, &cr);
  float c0 = 1.0f, s0 = 0.0f, ar = 0.0f, ai = 0.0f;
  for (int x = 0; x < HX_; ++x) {
    const float* p = st + (((size_t)bc * HX_ + x) * 9 + jy) * 2;
    float tr = p[0], ti = p[1];
    ar += tr * c0 - ti * s0;
    ai += tr * s0 + ti * c0;
    float nc = c0 * cr - s0 * sr;
    s0 = c0 * sr + s0 * cr; c0 = nc;
  }
  nft[(size_t)idx * 2] = ar * (1.0f / 65.0f);
  nft[(size_t)idx * 2 + 1] = ai * (1.0f / 65.0f);
}

__global__ void k_images(const float* __restrict__ nft, const float* __restrict__ w1,
                         const float* __restrict__ w2, float* __restrict__ img) {
  int idx = blockIdx.x * blockDim.x + threadIdx.x;
  if (idx >= B_ * S_ * 289) return;
  int py = idx % 17;
  int t = idx / 17;
  int px = t % 17;
  int o = (t / 17) % S_;
  int b = t / (17 * S_);
  float ar = 0.0f, ai = 0.0f;
  if (py != 8) {
    int conj = (py > 8);
    int qx = conj ? (16 - px) : px;
    int qy = conj ? (16 - py) : py; // 0..7
    for (int i = 0; i < WID_; ++i) {
      const float* f = nft + (((size_t)(b * WID_ + i) * 17 + qx) * 9 + qy) * 2;
      const float* w = w1 + ((((size_t)i * S_ + o) * 17 + qx) * 8 + qy) * 2;
      ar += f[0] * w[0] - f[1] * w[1];
      ai += f[0] * w[1] + f[1] * w[0];
    }
    if (conj) ai = -ai;
  } else {
    int conj = (px > 8);
    int qx = conj ? (16 - px) : px; // 0..8 (fx=-8..0)
    for (int i = 0; i < WID_; ++i) {
      const float* f = nft + (((size_t)(b * WID_ + i) * 17 + qx) * 9 + 8) * 2;
      const float* w = w2 + (((size_t)i * S_ + o) * 9 + qx) * 2;
      ar += f[0] * w[0] - f[1] * w[1];
      ai += f[0] * w[1] + f[1] * w[0];
    }
    if (conj) ai = -ai;
    if (px == 8) ai = 0.0f; // center forced real
  }
  img[(size_t)idx * 2] = ar;
  img[(size_t)idx * 2 + 1] = ai;
}

__global__ void k_seg(const int* __restrict__ ind, const int* __restrict__ sep,
                      int* __restrict__ seg) {
  int idx = blockIdx.x * blockDim.x + threadIdx.x;
  if (idx >= B_ * NY) return;
  int b = idx / NY, j = idx - b * NY;
  const int* sp = sep + b * (S_ + 1);
  int s = 0;
#pragma unroll
  for (int t = 1; t <= S_; ++t) if (sp[t] <= j) s = t;
  seg[b * NY + ind[idx]] = s;
}

__global__ void k_mminit(int* __restrict__ mm) {
  int i = blockIdx.x * blockDim.x + threadIdx.x;
  if (i >= B_ * S_) return;
  mm[i * 4 + 0] = 0x7F800000;
  mm[i * 4 + 1] = 0x7F800000;
  mm[i * 4 + 2] = 0;
  mm[i * 4 + 3] = 0;
}

__global__ void k_minmax(const float* __restrict__ loc, const int* __restrict__ seg,
                         int* __restrict__ mm) {
  int idx = blockIdx.x * blockDim.x + threadIdx.x;
  if (idx >= B_ * NY) return;
  int b = idx / NY;
  int s = seg[idx];
  int l0 = __float_as_int(loc[(size_t)idx * 2]);
  int l1 = __float_as_int(loc[(size_t)idx * 2 + 1]);
  int* p = mm + (b * S_ + s) * 4;
  atomicMin(&p[0], l0);
  atomicMin(&p[1], l1);
  atomicMax(&p[2], l0);
  atomicMax(&p[3], l1);
}

__global__ void k_nueval(const float* __restrict__ loc, const int* __restrict__ seg,
                         const int* __restrict__ mm, const float* __restrict__ img,
                         const float* __restrict__ nu_w, const float* __restrict__ nu_b,
                         float* __restrict__ xnu) {
  int idx = blockIdx.x * blockDim.x + threadIdx.x;
  if (idx >= B_ * NY) return;
  int b = idx / NY;
  int s = seg[idx];
  const int* p = mm + (b * S_ + s) * 4;
  float mn0 = __int_as_float(p[0]), mn1 = __int_as_float(p[1]);
  float mx0 = __int_as_float(p[2]), mx1 = __int_as_float(p[3]);
  float l0 = loc[(size_t)idx * 2], l1 = loc[(size_t)idx * 2 + 1];
  float om0 = (l0 - mn0) / (mx0 - mn0 + EPS_) * TWO_PI_F - PI_F;
  float om1 = (l1 - mn1) / (mx1 - mn1 + EPS_) * TWO_PI_F - PI_F;
  const float* im = img + ((size_t)(b * S_ + s)) * 289 * 2;
  float cx, sx, c8x, s8x, cy, sy, c8y, s8y;
  __sincosf(om0, &sx, &cx);
  __sincosf(-8.0f * om0, &s8x, &c8x);
  __sincosf(om1, &sy, &cy);
  __sincosf(-8.0f * om1, &s8y, &c8y);
  float eyc = c8y, eys = s8y;
  float acc = 0.0f;
  for (int jy = 0; jy < 17; ++jy) {
    float exc = c8x, exs = s8x;
    float tr = 0.0f, ti = 0.0f;
#pragma unroll
    for (int jx = 0; jx < 17; ++jx) {
      float ir = im[(jx * 17 + jy) * 2];
      float ii = im[(jx * 17 + jy) * 2 + 1];
      tr += ir * exc - ii * exs;
      ti += ir * exs + ii * exc;
      float nc = exc * cx - exs * sx;
      exs = exc * sx + exs * cx; exc = nc;
    }
    acc += tr * eyc - ti * eys;
    float nc = eyc * cy - eys * sy;
    eys = eyc * sy + eys * cy; eyc = nc;
  }
  float val = acc * (1.0f / 17.0f);
  xnu[idx] = val * nu_w[s] + nu_b[s];
}

// =====================================================================
// Host-side launcher
// =====================================================================
static inline int cdiv(int a, int b) { return (a + b - 1) / b; }

extern "C" void kernel_launch(void* const* d_in, const int* in_sizes, int n_in,
                              void* d_out, int out_size, void* d_ws, size_t ws_size,
                              hipStream_t stream) {
  (void)in_sizes; (void)n_in; (void)out_size; (void)ws_size;
  const float* x        = (const float*)d_in[0];
  const float* loc      = (const float*)d_in[1];
  const int*   ind      = (const int*)d_in[2];
  const int*   sep      = (const int*)d_in[3];
  const float* W_embed1 = (const float*)d_in[4];
  const float* b_embed1 = (const float*)d_in[5];
  const float* W_embed2 = (const float*)d_in[6];
  const float* b_embed2 = (const float*)d_in[7];
  const float* W_fc0    = (const float*)d_in[8];
  const float* b_fc0    = (const float*)d_in[9];
  const float* conv_w1  = (const float*)d_in[10];
  const float* conv_w2  = (const float*)d_in[11];
  const float* wpt      = (const float*)d_in[12];
  const float* wpt_b    = (const float*)d_in[13];
  const float* nu_w1    = (const float*)d_in[14];
  const float* nu_w2    = (const float*)d_in[15];
  const float* nu_w     = (const float*)d_in[16];
  const float* nu_b     = (const float*)d_in[17];
  const float* W_de3    = (const float*)d_in[18];
  const float* b_de3    = (const float*)d_in[19];
  const float* W_fc1    = (const float*)d_in[20];
  const float* b_fc1    = (const float*)d_in[21];
  const float* W_fc2    = (const float*)d_in[22];
  const float* b_fc2    = (const float*)d_in[23];
  float* out = (float*)d_out;
  float* ws  = (float*)d_ws;

  float* pxe  = ws + OFF_PXE;
  float* ple  = ws + OFF_PLE;
  float* h    = ws + OFF_H;
  float* hb   = ws + OFF_HB;
  float* st   = ws + OFF_ST;
  float* fr   = ws + OFF_FR;
  float* fo   = ws + OFF_FO;
  float* nft  = ws + OFF_NFT;
  float* img  = ws + OFF_IMG;
  float* hl   = ws + OFF_HL;
  float* xnu  = ws + OFF_XNU;
  int*   mm   = (int*)(ws + OFF_MNMX);
  int*   segp = (int*)(ws + OFF_SEG);
  float* pde  = ws + OFF_PDE;

  const int TB = 256;

  // --- embeddings (WMMA streaming GEMMs, K split 4x) ---
  k_gemm_xe<<<(G_ / 16) * KSPLIT * 32 / TB, TB, 0, stream>>>(x, W_embed1, pxe);
  k_gemm_le<<<(G_ / 16) * KSPLIT * 32 / TB, TB, 0, stream>>>(loc, W_embed2, ple);
  k_fc0<<<cdiv(B_ * G_, TB), TB, 0, stream>>>(pxe, ple, b_embed1, b_embed2, W_fc0, b_fc0, h);

  // --- 4 FNO layers ---
  for (int l = 0; l < 4; ++l) {
    k_dfty<<<cdiv(B_ * WID_ * HX_ * M1_, TB), TB, 0, stream>>>(h, st);
    k_dftx<<<cdiv(B_ * WID_ * 16 * M1_, TB), TB, 0, stream>>>(st, fr);
    k_mix<<<cdiv(B_ * WID_ * 16 * M1_, TB), TB, 0, stream>>>(fr, conv_w1, conv_w2, fo, l);
    k_idftx<<<cdiv(B_ * WID_ * HX_ * M1_, TB), TB, 0, stream>>>(fo, st);
    k_idfty<<<cdiv(B_ * WID_ * G_, TB), TB, 0, stream>>>(st, hb);
    k_wpt<<<(B_ * G_ / 16) * 32 / TB, TB, 0, stream>>>(h, hb, wpt, wpt_b, l, (l < 3) ? 1 : 0);
  }

  // --- local branch: fc1 -> gelu -> fc2 ---
  k_fc12<<<cdiv(B_ * G_, TB), TB, 0, stream>>>(h, W_fc1, b_fc1, W_fc2, b_fc2, hl);

  // --- NU spectral branch ---
  k_nudfty<<<cdiv(B_ * WID_ * HX_ * 9, TB), TB, 0, stream>>>(h, st);
  k_nudftx<<<cdiv(B_ * WID_ * 17 * 9, TB), TB, 0, stream>>>(st, nft);
  k_images<<<cdiv(B_ * S_ * 289, TB), TB, 0, stream>>>(nft, nu_w1, nu_w2, img);
  k_seg<<<cdiv(B_ * NY, TB), TB, 0, stream>>>(ind, sep, segp);
  k_mminit<<<1, 64, 0, stream>>>(mm);
  k_minmax<<<cdiv(B_ * NY, TB), TB, 0, stream>>>(loc, segp, mm);
  k_nueval<<<cdiv(B_ * NY, TB), TB, 0, stream>>>(loc, segp, mm, img, nu_w, nu_b, xnu);

  // --- decode: K-split WMMA GEMM then reduce (+bias +NU) into final output ---
  k_gemm_de3<<<(NY / 16) * KSPLIT * 32 / TB, TB, 0, stream>>>(hl, W_de3, pde);
  k_red_de3<<<cdiv(B_ * NY, TB), TB, 0, stream>>>(pde, b_de3, xnu, out);
}